// Convention_33182917329120
// MI455X (gfx1250) — compile-verified
//
#include <hip/hip_runtime.h>
#include <math.h>

typedef __attribute__((ext_vector_type(2))) float v2f;
typedef __attribute__((ext_vector_type(8))) float v8f;

#define DT 0.05f
#define NSTEPS 21

// gfx1250 has a hardware v_tanh_f32 (trans pipe, co-executes with VALU).
#if __has_builtin(__builtin_amdgcn_tanhf)
__device__ __forceinline__ float fast_tanh(float x) { return __builtin_amdgcn_tanhf(x); }
#else
__device__ __forceinline__ float fast_tanh(float x) { return tanhf(x); }
#endif

// One wave = 16 tasks (tasks -> B-matrix columns / lanes 0-15).
// Layer-1 of both MLPs runs on the matrix pipe via V_WMMA_F32_16X16X4_F32:
//   D = A(weights, 16x4 padded) * B(states, 4x16) + C(per-lane affine part).
// All constant/per-lane affine contributions (biases, s_star part, and the
// robot net's Wr1[:,2]*z term) are folded into C, so the live K dim is just
// the evolving 2-state -- and both WMMAs share one B operand with zero
// cross-lane data movement (states already sit in the correct lanes for the
// B K-striping).
__global__ __launch_bounds__(256) void rollout_kernel(
    const float* __restrict__ omega, const float* __restrict__ s0,
    const float* __restrict__ Wh1, const float* __restrict__ bh1,
    const float* __restrict__ Wh2, const float* __restrict__ bh2,
    const float* __restrict__ Wr1, const float* __restrict__ br1,
    const float* __restrict__ Wr2, const float* __restrict__ br2,
    float* __restrict__ partial, int nTasks)
{
  const int lane = threadIdx.x & 31;
  const int wave = threadIdx.x >> 5;
  const int m    = lane & 15;                 // row/col index within the tile
  const int taskRaw = blockIdx.x * 128 + wave * 16 + m;
  const bool valid  = (taskRaw < nTasks) && (lane < 16);
  const int  task   = (taskRaw < nTasks) ? taskRaw : (nTasks - 1);

  // targets (lanes 16-31 duplicate lanes 0-15; their results are masked out)
  const float t0 = omega[task];
  const float t1 = omega[nTasks + task];

  // uniform (scalar) weights
  const float wh2_0 = Wh2[0], wh2_1 = Wh2[1], wh2_2 = Wh2[2], wh2_3 = Wh2[3];
  const float bh2s  = bh2[0];
  const float wr2_0 = Wr2[0], wr2_1 = Wr2[1], wr2_2 = Wr2[2];
  const float br2s  = br2[0];
  const float wr1_02 = Wr1[2], wr1_12 = Wr1[5], wr1_22 = Wr1[8];  // Wr1[:,2]
  const float br1_0 = br1[0], br1_1 = br1[1], br1_2 = br1[2];

  // ---- A operand, human net: rows 0-3 = Wh1[:,2:4] in K0/K1; K2,K3 = 0 ----
  // A layout (16x4 f32): VGPR0 = K0 (lanes0-15) | K2 (lanes16-31); VGPR1 = K1 | K3.
  const int rh = (m < 4) ? m : 0;
  const float ahw2 = Wh1[rh * 4 + 2], ahw3 = Wh1[rh * 4 + 3];
  v2f Ah;
  Ah.x = (lane < 4) ? ahw2 : 0.f;
  Ah.y = (lane < 4) ? ahw3 : 0.f;

  // ---- C operand, human net: c_h[j] = bh1[j] + Wh1[j,0]*t0 + Wh1[j,1]*t1 ----
  v8f Ch;
  Ch[0] = fmaf(Wh1[0], t0, fmaf(Wh1[1], t1, bh1[0]));
  Ch[1] = fmaf(Wh1[4], t0, fmaf(Wh1[5], t1, bh1[1]));
  Ch[2] = fmaf(Wh1[8], t0, fmaf(Wh1[9], t1, bh1[2]));
  Ch[3] = fmaf(Wh1[12], t0, fmaf(Wh1[13], t1, bh1[3]));
  Ch[4] = 0.f; Ch[5] = 0.f; Ch[6] = 0.f; Ch[7] = 0.f;

  // ---- A operand, robot net: rows 0-2 = Wr1[:,0:2] in K0/K1; K2,K3 = 0 ----
  const int rr = (m < 3) ? m : 0;
  const float arw0 = Wr1[rr * 3 + 0], arw1 = Wr1[rr * 3 + 1];
  v2f Ar;
  Ar.x = (lane < 3) ? arw0 : 0.f;
  Ar.y = (lane < 3) ? arw1 : 0.f;

  float ss0 = s0[0], ss1 = s0[1];
  float err = 0.f, eff = 0.f;

  for (int step = 0; step < NSTEPS - 1; ++step) {
    // tracking cost with current state
    float e0 = t0 - ss0, e1 = t1 - ss1;
    err = fmaf(10.f * e0, e0, fmaf(e1, e1, err));

    // shared B operand (states already lane-resident for the K-striping)
    v2f B;
    B.x = (lane < 16) ? ss0 : 0.f;   // K0 | K2(=0, matches zero A columns)
    B.y = (lane < 16) ? ss1 : 0.f;   // K1 | K3(=0)

    // human net layer 1 on the matrix pipe
    v8f Dh = __builtin_amdgcn_wmma_f32_16x16x4_f32(
        false, Ah, false, B, (short)0, Ch, false, false);
    float h0 = fast_tanh(Dh[0]);
    float h1 = fast_tanh(Dh[1]);
    float h2 = fast_tanh(Dh[2]);
    float h3 = fast_tanh(Dh[3]);
    float z = fmaf(wh2_0, h0, fmaf(wh2_1, h1, fmaf(wh2_2, h2, fmaf(wh2_3, h3, bh2s))));
    eff = fmaf(z, z, eff);

    // robot net layer 1: the Wr1[:,2]*z term rides in the C accumulator
    // (per-lane fmacs), so no cross-lane move of z is needed at all.
    v8f Cr;
    Cr[0] = fmaf(wr1_02, z, br1_0);
    Cr[1] = fmaf(wr1_12, z, br1_1);
    Cr[2] = fmaf(wr1_22, z, br1_2);
    Cr[3] = 0.f; Cr[4] = 0.f; Cr[5] = 0.f; Cr[6] = 0.f; Cr[7] = 0.f;
    v8f Dr = __builtin_amdgcn_wmma_f32_16x16x4_f32(
        false, Ar, false, B, (short)0, Cr, false, false);
    float r0 = fast_tanh(Dr[0]);
    float r1 = fast_tanh(Dr[1]);
    float r2 = fast_tanh(Dr[2]);
    float a = fmaf(wr2_0, r0, fmaf(wr2_1, r1, fmaf(wr2_2, r2, br2s)));

    // s_next = [s0 + dt*s1, s1 + dt*a]
    float n0 = fmaf(DT, ss1, ss0);
    ss1 = fmaf(DT, a, ss1);
    ss0 = n0;
  }

  // terminal cost
  float e0 = t0 - ss0, e1 = t1 - ss1;
  err = fmaf(10.f * e0, e0, fmaf(e1, e1, err));

  float cost = valid ? (err + eff) : 0.f;

  // deterministic wave reduction (upper lanes already zeroed)
  #pragma unroll
  for (int off = 16; off > 0; off >>= 1)
    cost += __shfl_xor(cost, off, 32);

  __shared__ float sh[8];
  if (lane == 0) sh[wave] = cost;
  __syncthreads();
  if (threadIdx.x == 0) {
    float s = 0.f;
    #pragma unroll
    for (int w = 0; w < 8; ++w) s += sh[w];
    partial[blockIdx.x] = s;
  }
}

// Single-block, fixed-order final reduction (deterministic, no atomics).
__global__ __launch_bounds__(256) void reduce_kernel(
    const float* __restrict__ partial, int n, float invN, float* __restrict__ out)
{
  __shared__ float sh[256];
  float s = 0.f;
  for (int i = threadIdx.x; i < n; i += 256) s += partial[i];
  sh[threadIdx.x] = s;
  __syncthreads();
  for (int off = 128; off > 0; off >>= 1) {
    if ((int)threadIdx.x < off) sh[threadIdx.x] += sh[threadIdx.x + off];
    __syncthreads();
  }
  if (threadIdx.x == 0) out[0] = sh[0] * invN;
}

extern "C" void kernel_launch(void* const* d_in, const int* in_sizes, int n_in,
                              void* d_out, int out_size, void* d_ws, size_t ws_size,
                              hipStream_t stream) {
  const float* omega = (const float*)d_in[0];
  const float* s0    = (const float*)d_in[1];
  const float* Wh1   = (const float*)d_in[2];
  const float* bh1   = (const float*)d_in[3];
  const float* Wh2   = (const float*)d_in[4];
  const float* bh2   = (const float*)d_in[5];
  const float* Wr1   = (const float*)d_in[6];
  const float* br1   = (const float*)d_in[7];
  const float* Wr2   = (const float*)d_in[8];
  const float* br2   = (const float*)d_in[9];

  const int nTasks = in_sizes[0] / 2;          // omega is [2, N]
  const int blocks = (nTasks + 127) / 128;     // 128 tasks per 256-thread block
  float* partial = (float*)d_ws;               // blocks * 4 bytes of scratch

  rollout_kernel<<<blocks, 256, 0, stream>>>(
      omega, s0, Wh1, bh1, Wh2, bh2, Wr1, br1, Wr2, br2, partial, nTasks);
  reduce_kernel<<<1, 256, 0, stream>>>(
      partial, blocks, 1.0f / (float)nTasks, (float*)d_out);
}